// GCNBlock_10282151707323
// MI455X (gfx1250) — compile-verified
//
#include <hip/hip_runtime.h>
#include <hip/hip_bf16.h>
#include <stdint.h>

typedef __attribute__((ext_vector_type(2))) float v2f;
typedef __attribute__((ext_vector_type(8))) float v8f;

#define IN_F  256
#define OUT_F 256
#define LDS_PITCH 258   // 256 + 2 pad: stride 258 % 64 banks = 2 -> conflict-free column reads

// ---------------------------------------------------------------------------
// 1) init: out[idx] = conv_bias[col]; deg[n] = 1 (self loop); accum = 0
// ---------------------------------------------------------------------------
__global__ __launch_bounds__(256) void init_kernel(float* __restrict__ out,
                                                   const float* __restrict__ bias,
                                                   float* __restrict__ deg,
                                                   float* __restrict__ accum,
                                                   int total, int n) {
    int idx = blockIdx.x * 256 + threadIdx.x;
    if (idx < total) out[idx] = bias[idx & (OUT_F - 1)];
    if (idx < n)     deg[idx] = 1.0f;
    if (idx < 2)     accum[idx] = 0.0f;
}

// ---------------------------------------------------------------------------
// 2) degree: deg[dst[e]] += 1   (global_atomic_add_f32, non-returning)
// ---------------------------------------------------------------------------
__global__ __launch_bounds__(256) void deg_kernel(const int* __restrict__ dst,
                                                  float* __restrict__ deg, int E) {
    int e = blockIdx.x * 256 + threadIdx.x;
    if (e < E) unsafeAtomicAdd(&deg[dst[e]], 1.0f);
}

// ---------------------------------------------------------------------------
// 3) dinv[n] = rsqrt(deg[n])  (in place)
// ---------------------------------------------------------------------------
__global__ __launch_bounds__(256) void dinv_kernel(float* __restrict__ deg, int n) {
    int i = blockIdx.x * 256 + threadIdx.x;
    if (i < n) deg[i] = rsqrtf(deg[i]);
}

// ---------------------------------------------------------------------------
// 4a) pack W rows into float2 pairs: Wp[p*256+n] = {W[2p][n], W[2p+1][n]}
//     so a B fragment of V_WMMA_F32_16X16X4_F32 is one coalesced b64 load.
// ---------------------------------------------------------------------------
__global__ __launch_bounds__(256) void packW_kernel(const float* __restrict__ W,
                                                    float2* __restrict__ Wp) {
    int idx = blockIdx.x * 256 + threadIdx.x;       // 0 .. 128*256-1
    int p = idx >> 8;
    int n = idx & 255;
    Wp[idx] = make_float2(W[(size_t)(2 * p) * OUT_F + n],
                          W[(size_t)(2 * p + 1) * OUT_F + n]);
}

// ---------------------------------------------------------------------------
// 4b) h = x @ W.  Block = 8 waves computes a 64x256 strip.
//     - x strip (64x256 f32) staged in LDS via global_load_async_to_lds_b128
//       (ASYNCcnt-tracked CDNA5 async copy), padded pitch kills bank conflicts.
//     - wave w: M-tile mi = w>>1, N-half nh = w&1 (8 N-tiles, 64 acc VGPRs);
//       one ds_load_b64 A-fragment feeds 8 v_wmma_f32_16x16x4_f32.
// ---------------------------------------------------------------------------
__global__ __launch_bounds__(256) void gemm_wmma_kernel(const float* __restrict__ x,
                                                        const float2* __restrict__ Wp,
                                                        float* __restrict__ h,
                                                        int N) {
    __shared__ float xs[64 * LDS_PITCH];            // 66,048 B

    const int t = threadIdx.x;
    const int stripBase = blockIdx.x * 64;          // first row of this strip

    // ---- async stage: 64 rows x 256 cols of x -> LDS (clamped tail rows) ----
    const uint32_t xsBase = (uint32_t)(uintptr_t)&xs[0];
#pragma unroll
    for (int i = 0; i < 16; ++i) {
        int f   = t + i * 256;                      // float4 index within strip
        int row = f >> 6;                           // 64 float4 per row
        int c4  = (f & 63) << 2;                    // column (floats)
        int gRow = stripBase + row;
        if (gRow > N - 1) gRow = N - 1;             // clamp (stores are guarded)
        const float* gp = x + (size_t)gRow * IN_F + c4;
        uint32_t lds = xsBase + (uint32_t)(row * LDS_PITCH + c4) * 4u;
        asm volatile("global_load_async_to_lds_b128 %0, %1, off"
                     :: "v"(lds), "v"(gp) : "memory");
    }
    asm volatile("s_wait_asynccnt 0x0" ::: "memory");
    __syncthreads();

    // ---- per-wave tile assignment ----
    const int lane = t & 31;
    const int half = lane >> 4;                     // K pair select
    const int l16  = lane & 15;
    const int w    = t >> 5;                        // 0..7
    const int mi   = w >> 1;                        // 0..3  (16-row M tile)
    const int nh   = w & 1;                         // 0..1  (128-col N half)

    v8f acc[8];
#pragma unroll
    for (int nt = 0; nt < 8; ++nt) acc[nt] = (v8f){0.f,0.f,0.f,0.f,0.f,0.f,0.f,0.f};

    const float* arow = &xs[(mi * 16 + l16) * LDS_PITCH];
    const float2* wb  = Wp + nh * 128 + l16;

#pragma unroll 2
    for (int k = 0; k < IN_F; k += 4) {
        const float2 av = *(const float2*)&arow[k + 2 * half];   // ds_load_b64
        v2f a; a.x = av.x; a.y = av.y;
        const float2* wrow = wb + (size_t)((k >> 1) + half) * OUT_F;
#pragma unroll
        for (int nt = 0; nt < 8; ++nt) {
            float2 bw = wrow[nt * 16];                           // global_load_b64
            v2f b; b.x = bw.x; b.y = bw.y;
            acc[nt] = __builtin_amdgcn_wmma_f32_16x16x4_f32(false, a, false, b,
                                                            (short)0, acc[nt],
                                                            false, false);
        }
    }

    // ---- store (N % 16 == 0 so validity is uniform per M tile) ----
    if (stripBase + mi * 16 < N) {
        float* hb = h + (size_t)(stripBase + mi * 16) * OUT_F + nh * 128 + l16;
#pragma unroll
        for (int nt = 0; nt < 8; ++nt) {
            float* hc = hb + nt * 16;
#pragma unroll
            for (int v = 0; v < 8; ++v)
                hc[(size_t)(v + 8 * half) * OUT_F] = acc[nt][v];
        }
    }
}

// ---------------------------------------------------------------------------
// 5) edge aggregation: out[dst] += h[src] * (dinv[src]*dinv[dst])
//    64 lanes per edge, float4 gather, global_atomic_add_f32 scatter.
//    h and out both live in the 192MB L2 -> runs at L2 bandwidth.
// ---------------------------------------------------------------------------
__global__ __launch_bounds__(256) void edge_agg_kernel(const int* __restrict__ src,
                                                       const int* __restrict__ dst,
                                                       const float* __restrict__ dinv,
                                                       const float* __restrict__ h,
                                                       float* __restrict__ out, int E) {
    int t = blockIdx.x * 256 + threadIdx.x;
    int e = t >> 6;                       // 64 threads per edge
    if (e >= E) return;
    int c4 = (t & 63) << 2;               // feature offset (x4 floats)
    int s = src[e], d = dst[e];
    float norm = dinv[s] * dinv[d];
    const float4 hv = *reinterpret_cast<const float4*>(h + (size_t)s * OUT_F + c4);
    float* o = out + (size_t)d * OUT_F + c4;
    unsafeAtomicAdd(o + 0, hv.x * norm);
    unsafeAtomicAdd(o + 1, hv.y * norm);
    unsafeAtomicAdd(o + 2, hv.z * norm);
    unsafeAtomicAdd(o + 3, hv.w * norm);
}

// ---------------------------------------------------------------------------
// 6) add self-loop term, accumulate sum / sum-of-squares for LayerNorm
// ---------------------------------------------------------------------------
__global__ __launch_bounds__(256) void finalize_sum_kernel(float* __restrict__ out,
                                                           const float* __restrict__ h,
                                                           const float* __restrict__ dinv,
                                                           float* __restrict__ accum,
                                                           int total) {
    int idx = blockIdx.x * 256 + threadIdx.x;
    float v = 0.f, v2 = 0.f;
    if (idx < total) {
        int row = idx >> 8;               // OUT_F == 256
        float di = dinv[row];
        float val = out[idx] + h[idx] * di * di;   // self-loop: norm = 1/deg
        out[idx] = val;
        v = val; v2 = val * val;
    }
#pragma unroll
    for (int off = 16; off > 0; off >>= 1) {       // wave32 reduction
        v  += __shfl_down(v,  off, 32);
        v2 += __shfl_down(v2, off, 32);
    }
    __shared__ float s1[8], s2[8];
    int wid = threadIdx.x >> 5;
    if ((threadIdx.x & 31) == 0) { s1[wid] = v; s2[wid] = v2; }
    __syncthreads();
    if (threadIdx.x < 8) {
        v = s1[threadIdx.x]; v2 = s2[threadIdx.x];
#pragma unroll
        for (int off = 4; off > 0; off >>= 1) {
            v  += __shfl_down(v,  off, 32);
            v2 += __shfl_down(v2, off, 32);
        }
        if (threadIdx.x == 0) {
            unsafeAtomicAdd(&accum[0], v);
            unsafeAtomicAdd(&accum[1], v2);
        }
    }
}

// ---------------------------------------------------------------------------
// 7) LayerNorm (graph mode: global mean/std) + PReLU, in place on out
// ---------------------------------------------------------------------------
__global__ __launch_bounds__(256) void ln_prelu_kernel(float* __restrict__ out,
                                                       const float* __restrict__ accum,
                                                       const float* __restrict__ lnw,
                                                       const float* __restrict__ lnb,
                                                       const float* __restrict__ pa,
                                                       int total) {
    int idx = blockIdx.x * 256 + threadIdx.x;
    if (idx >= total) return;
    float M   = (float)total;
    float mu  = accum[0] / M;
    float var = accum[1] / M - mu * mu;
    float stdv = sqrtf(fmaxf(var, 0.f));
    float y = (out[idx] - mu) / (stdv + 1e-5f);
    int col = idx & (OUT_F - 1);
    y = y * lnw[col] + lnb[col];
    out[idx] = (y >= 0.f) ? y : pa[0] * y;
}

// ---------------------------------------------------------------------------
extern "C" void kernel_launch(void* const* d_in, const int* in_sizes, int n_in,
                              void* d_out, int out_size, void* d_ws, size_t ws_size,
                              hipStream_t stream) {
    const float* x     = (const float*)d_in[0];
    const int*   eidx  = (const int*)d_in[1];     // (2, E) flattened
    const float* W     = (const float*)d_in[3];
    const float* bias  = (const float*)d_in[4];
    const float* lnw   = (const float*)d_in[5];
    const float* lnb   = (const float*)d_in[6];
    const float* pa    = (const float*)d_in[7];

    const int N = in_sizes[0] / IN_F;
    const int E = in_sizes[1] / 2;
    const int total = N * OUT_F;

    const int* src = eidx;
    const int* dst = eidx + E;

    float*  out   = (float*)d_out;
    float*  h     = (float*)d_ws;                       // N*256 floats
    float*  deg   = h + (size_t)N * OUT_F;              // N floats (becomes dinv)
    float*  accum = deg + N;                            // 2 floats
    float2* Wp    = (float2*)(accum + 2);               // 128*256 float2 (256 KB)

    const int eltBlocks = (total + 255) / 256;

    init_kernel<<<eltBlocks, 256, 0, stream>>>(out, bias, deg, accum, total, N);
    deg_kernel<<<(E + 255) / 256, 256, 0, stream>>>(dst, deg, E);
    dinv_kernel<<<(N + 255) / 256, 256, 0, stream>>>(deg, N);

    packW_kernel<<<(IN_F / 2 * OUT_F) / 256, 256, 0, stream>>>(W, Wp);

    const int strips = (N + 63) / 64;                   // 782
    gemm_wmma_kernel<<<strips, 256, 0, stream>>>(x, Wp, h, N);

    const long long eThreads = (long long)E * 64;
    edge_agg_kernel<<<(int)((eThreads + 255) / 256), 256, 0, stream>>>(src, dst, deg, h, out, E);

    finalize_sum_kernel<<<eltBlocks, 256, 0, stream>>>(out, h, deg, accum, total);
    ln_prelu_kernel<<<eltBlocks, 256, 0, stream>>>(out, accum, lnw, lnb, pa, total);
}